// MyModel_61933428412709
// MI455X (gfx1250) — compile-verified
//
#include <hip/hip_runtime.h>
#include <math.h>

typedef __attribute__((ext_vector_type(2))) float v2f;
typedef __attribute__((ext_vector_type(8))) float v8f;

#define N 4096
#define NB (N / 16)          // 256 row-tiles
#define VLEN (N * 16)        // 65536 floats per block-vector
#define SPLITK 8
#define KCH (N / SPLITK)     // 512 K per partial wave

// ---------------------------------------------------------------------------
// Init: V = block of 16 identical columns, mildly varying by row (generic
// overlap with every eigenvector).
// ---------------------------------------------------------------------------
__global__ void init_V(float* __restrict__ V) {
    int i = blockIdx.x * blockDim.x + threadIdx.x;   // 0..VLEN-1
    int row = i >> 4;
    V[i] = 1.0f + 0.001f * (float)(row & 63);
}

__global__ void init_scal(float* __restrict__ scal) {
    if (threadIdx.x == 0) { scal[0] = 0.0f; scal[1] = 0.0f; scal[2] = 1.0f; scal[3] = 0.0f; }
}

// ---------------------------------------------------------------------------
// Partial GEMM: Wp[p] = S[tile rows, Kchunk p] * V[Kchunk p]
// S = x + x^T built on the fly. One wave (32 lanes) per (tile, p) pair,
// chaining V_WMMA_F32_16X16X4_F32 over its 512-wide K chunk.
//
// fp32 WMMA lane layout (ISA 7.12.2):
//   A 16x4 : lanes 0-15 -> M=lane, K={0,1} in vgpr{0,1}; lanes 16-31 -> K={2,3}
//   B 4x16 : vgpr0 holds rows K=0 (lanes 0-15) / K=2 (lanes 16-31); vgpr1 K=1/K=3
//   D      : vgpr r -> (M = half*8 + r, N = lane&15)
// ---------------------------------------------------------------------------
__global__ __launch_bounds__(32) void sym_gemm_partial(const float* __restrict__ x,
                                                       const float* __restrict__ V,
                                                       float* __restrict__ Wp) {
    const int tile = blockIdx.x & (NB - 1);   // 0..NB-1   (row tile)
    const int part = blockIdx.x >> 8;         // 0..SPLITK-1 (K chunk)
    const int i0   = tile * 16;
    const int ks   = part * KCH;
    const int lane = threadIdx.x;             // 0..31
    const int half = lane >> 4;
    const int l    = lane & 15;

    const float* xrow = x + (size_t)(i0 + l) * N;   // x[i0+l, :]
    v8f acc = {};

#pragma unroll 8
    for (int k = ks; k < ks + KCH; k += 4) {
        const int kk0 = k + half * 2;
        const int kk1 = kk0 + 1;

        // A[m, kk] = S[i0+m, kk] = x[i0+m, kk] + x[kk, i0+m]
        const v2f arow = *reinterpret_cast<const v2f*>(xrow + kk0);  // b64 load
        v2f a, b;
        a.x = arow.x + x[(size_t)kk0 * N + (i0 + l)];
        a.y = arow.y + x[(size_t)kk1 * N + (i0 + l)];
        // B[kk, n] = V[kk, n]
        b.x = V[kk0 * 16 + l];
        b.y = V[kk1 * 16 + l];

        // 8 args: (neg_a, A, neg_b, B, c_mod, C, reuse_a, reuse_b)
        acc = __builtin_amdgcn_wmma_f32_16x16x4_f32(false, a, false, b,
                                                    (short)0, acc, false, false);
    }

    float* wout = Wp + (size_t)part * VLEN;
#pragma unroll
    for (int r = 0; r < 8; ++r) {
        const int row = i0 + half * 8 + r;
        wout[row * 16 + l] = acc[r];
    }
}

// ---------------------------------------------------------------------------
// Deterministic split-K combine:
//   W = sum_p Wp[p]                      (shifted == 0)
//   W = sigma*V - sum_p Wp[p]            (shifted == 1, sigma = scal[0])
// Fixed summation order -> bitwise-identical across graph replays.
// ---------------------------------------------------------------------------
__global__ __launch_bounds__(256) void combine_splitk(const float* __restrict__ Wp,
                                                      const float* __restrict__ V,
                                                      float* __restrict__ W,
                                                      const float* __restrict__ scal,
                                                      int shifted) {
    const int i = blockIdx.x * 256 + threadIdx.x;   // 0..VLEN-1
    float s = 0.0f;
#pragma unroll
    for (int p = 0; p < SPLITK; ++p) s += Wp[(size_t)p * VLEN + i];
    if (shifted) s = scal[0] * V[i] - s;
    W[i] = s;
}

// ---------------------------------------------------------------------------
// Single-workgroup: mu = <V[:,0], W[:,0]> (Rayleigh quotient, V was unit),
// then V = W / ||W[:,0]||.  Optionally sets sigma = 1.5*|mu|.
// ---------------------------------------------------------------------------
__global__ __launch_bounds__(256) void reduce_normalize(float* __restrict__ V,
                                                        const float* __restrict__ W,
                                                        float* __restrict__ scal,
                                                        int setSigma) {
    __shared__ float s_vw[256];
    __shared__ float s_ww[256];
    const int t = threadIdx.x;

    float vw = 0.0f, ww = 0.0f;
    for (int i = t; i < N; i += 256) {
        float w = W[i * 16];          // column 0
        float v = V[i * 16];
        vw += v * w;
        ww += w * w;
    }
    s_vw[t] = vw;
    s_ww[t] = ww;
    __syncthreads();
    for (int s = 128; s > 0; s >>= 1) {
        if (t < s) { s_vw[t] += s_vw[t + s]; s_ww[t] += s_ww[t + s]; }
        __syncthreads();
    }
    if (t == 0) {
        const float mu = s_vw[0];
        scal[1] = mu;                 // Rayleigh quotient of previous unit V
        scal[2] = s_ww[0];
        if (setSigma) scal[0] = 1.5f * fabsf(mu);
    }
    const float inv = rsqrtf(s_ww[0]);
    for (int i = t; i < VLEN; i += 256) V[i] = W[i] * inv;
}

// lambda_min = sigma - mu   (mu = Rayleigh quotient of sigma*I - S)
__global__ void finish(float* __restrict__ out, const float* __restrict__ scal) {
    if (threadIdx.x == 0) out[0] = scal[0] - scal[1];
}

// ---------------------------------------------------------------------------
extern "C" void kernel_launch(void* const* d_in, const int* in_sizes, int n_in,
                              void* d_out, int out_size, void* d_ws, size_t ws_size,
                              hipStream_t stream) {
    (void)in_sizes; (void)n_in; (void)out_size;

    const size_t need = (size_t)((SPLITK + 2) * VLEN + 16) * sizeof(float);
    if (ws_size < need) return;

    const float* x = (const float*)d_in[0];
    float* out  = (float*)d_out;
    float* Wp   = (float*)d_ws;                 // SPLITK partial block-vectors
    float* V    = Wp + (size_t)SPLITK * VLEN;
    float* W    = V + VLEN;
    float* scal = W + VLEN;

    init_V<<<VLEN / 256, 256, 0, stream>>>(V);
    init_scal<<<1, 32, 0, stream>>>(scal);

    // Phase A: estimate spectral radius of S (sigma = 1.5 * |Rayleigh|).
    const int T1 = 40;
    for (int it = 0; it < T1; ++it) {
        sym_gemm_partial<<<NB * SPLITK, 32, 0, stream>>>(x, V, Wp);
        combine_splitk<<<VLEN / 256, 256, 0, stream>>>(Wp, V, W, scal, 0);
        reduce_normalize<<<1, 256, 0, stream>>>(V, W, scal, (it == T1 - 1) ? 1 : 0);
    }

    // Phase B: shifted power iteration on sigma*I - S -> eigvec of lambda_min.
    const int T2 = 400;
    for (int it = 0; it < T2; ++it) {
        sym_gemm_partial<<<NB * SPLITK, 32, 0, stream>>>(x, V, Wp);
        combine_splitk<<<VLEN / 256, 256, 0, stream>>>(Wp, V, W, scal, 1);
        reduce_normalize<<<1, 256, 0, stream>>>(V, W, scal, 0);
    }

    finish<<<1, 32, 0, stream>>>(out, scal);
}